// Model_83141976916149
// MI455X (gfx1250) — compile-verified
//
#include <hip/hip_runtime.h>

typedef __attribute__((ext_vector_type(2))) float v2f;
typedef __attribute__((ext_vector_type(8))) float v8f;

#define C_IN 64
#define C_OUT 32
#define IMG_H 224
#define IMG_W 224
#define HW (IMG_H * IMG_W)
#define OH 112
#define OW 112
#define WTILES 14                      // 224 / 16
#define TILES_PER_N (WTILES * OH)      // 1568
#define NBATCH 32
#define NTILES (NBATCH * TILES_PER_N)  // 50176
#define WAVES_PER_BLOCK 8

// Tiny pre-kernel: fold BN into per-channel scale/bias in workspace.
__global__ void bn_coef_kernel(const float* __restrict__ gamma,
                               const float* __restrict__ beta,
                               const float* __restrict__ mean,
                               const float* __restrict__ var,
                               float* __restrict__ sb) {
  int c = threadIdx.x;
  if (c < C_IN) {
    float s = gamma[c] * rsqrtf(var[c] + 1e-5f);
    sb[c] = s;
    sb[C_IN + c] = beta[c] - mean[c] * s;
  }
}

// Fused BN + ReLU + 1x1 conv (WMMA f32 16x16x4) + 2x2 avgpool.
// One wave handles: one n, rows (h, h+1), 16 consecutive w, all 32 out channels.
__global__ __launch_bounds__(256) void fused_bn_conv_pool(
    const float* __restrict__ x, const float* __restrict__ w,
    const float* __restrict__ sb, float* __restrict__ out) {
  const int lane = threadIdx.x & 31;
  const int wave = threadIdx.x >> 5;
  const int tile = blockIdx.x * WAVES_PER_BLOCK + wave;

  const int wb = tile % WTILES;
  const int hp = (tile / WTILES) % OH;
  const int n  = tile / TILES_PER_N;

  const int lm   = lane & 15;
  const int half = lane >> 4;
  const int h     = hp * 2;
  const int wbase = wb * 16;

  // Per-lane base: pixel column (N dim) = lm
  const float* xp  = x + (size_t)n * (C_IN * HW) + (size_t)h * IMG_W + wbase + lm;
  // A-matrix rows: M dim = lm; two M-tiles (out channels 0-15 and 16-31)
  const float* wp0 = w + lm * C_IN;
  const float* wp1 = w + (16 + lm) * C_IN;

  v8f dT0 = {}, dT1 = {}, dB0 = {}, dB1 = {};

#pragma unroll 4
  for (int kk = 0; kk < 16; ++kk) {
    const int k  = kk * 4;
    const int c0 = k + half * 2;  // this lane supplies K = c0, c0+1

    // BN coefficients for channels c0, c0+1 (cache-resident 512B table)
    v2f sv = *(const v2f*)(sb + c0);
    v2f bv = *(const v2f*)(sb + C_IN + c0);
    // A tiles: conv_w[o][c0], conv_w[o][c0+1]
    v2f a0 = *(const v2f*)(wp0 + c0);
    v2f a1 = *(const v2f*)(wp1 + c0);

    // Streamed x loads (each element read exactly once kernel-wide)
    const size_t co = (size_t)c0 * HW;
    float xt0 = __builtin_nontemporal_load(xp + co);
    float xt1 = __builtin_nontemporal_load(xp + co + HW);
    float xb0 = __builtin_nontemporal_load(xp + co + IMG_W);
    float xb1 = __builtin_nontemporal_load(xp + co + HW + IMG_W);

    v2f bT, bB;
    bT.x = fmaxf(fmaf(xt0, sv.x, bv.x), 0.0f);
    bT.y = fmaxf(fmaf(xt1, sv.y, bv.y), 0.0f);
    bB.x = fmaxf(fmaf(xb0, sv.x, bv.x), 0.0f);
    bB.y = fmaxf(fmaf(xb1, sv.y, bv.y), 0.0f);

    dT0 = __builtin_amdgcn_wmma_f32_16x16x4_f32(false, a0, false, bT, (short)0, dT0, false, false);
    dT1 = __builtin_amdgcn_wmma_f32_16x16x4_f32(false, a1, false, bT, (short)0, dT1, false, false);
    dB0 = __builtin_amdgcn_wmma_f32_16x16x4_f32(false, a0, false, bB, (short)0, dB0, false, false);
    dB1 = __builtin_amdgcn_wmma_f32_16x16x4_f32(false, a1, false, bB, (short)0, dB1, false, false);
  }

  // 2x2 avgpool: vertical add in registers, horizontal via lane shuffle (wave32)
  v8f s0, s1;
#pragma unroll
  for (int r = 0; r < 8; ++r) {
    float v0 = dT0[r] + dB0[r];
    float v1 = dT1[r] + dB1[r];
    s0[r] = (v0 + __shfl_xor(v0, 1, 32)) * 0.25f;
    s1[r] = (v1 + __shfl_xor(v1, 1, 32)) * 0.25f;
  }

  if ((lm & 1) == 0) {
    const int wo = wb * 8 + (lm >> 1);
    float* op = out + (((size_t)n * C_OUT) * OH + hp) * OW + wo;
#pragma unroll
    for (int r = 0; r < 8; ++r) {
      const int o0 = r + half * 8;        // D layout: VGPR r, lane-half -> M
      const int o1 = 16 + r + half * 8;
      op[(size_t)o0 * (OH * OW)] = s0[r];
      op[(size_t)o1 * (OH * OW)] = s1[r];
    }
  }
}

extern "C" void kernel_launch(void* const* d_in, const int* in_sizes, int n_in,
                              void* d_out, int out_size, void* d_ws, size_t ws_size,
                              hipStream_t stream) {
  const float* x      = (const float*)d_in[0];
  const float* conv_w = (const float*)d_in[1];
  const float* gamma  = (const float*)d_in[2];
  const float* beta   = (const float*)d_in[3];
  const float* rmean  = (const float*)d_in[4];
  const float* rvar   = (const float*)d_in[5];
  float* out = (float*)d_out;
  float* sb  = (float*)d_ws;  // 128 floats: scale[64] | bias[64]

  bn_coef_kernel<<<1, 64, 0, stream>>>(gamma, beta, rmean, rvar, sb);
  fused_bn_conv_pool<<<NTILES / WAVES_PER_BLOCK, 256, 0, stream>>>(x, conv_w, sb, out);
}